// HAN_45337674776785
// MI455X (gfx1250) — compile-verified
//
#include <hip/hip_runtime.h>
#include <math.h>

typedef float v2f __attribute__((ext_vector_type(2)));
typedef float v8f __attribute__((ext_vector_type(8)));

constexpr int N       = 10000;
constexpr int E       = 160000;
constexpr int IN_DIM  = 128;
constexpr int HEADS   = 8;
constexpr int HID     = 64;
constexpr int F       = 512;      // HEADS*HID
constexpr int SEM_HID = 128;
constexpr int OUT_DIM = 16;
constexpr int MP      = 3;

// ---------------------------------------------------------------- utilities

__global__ void zero_f_kernel(float* __restrict__ p, int n) {
    int i = blockIdx.x * blockDim.x + threadIdx.x;
    if (i < n) p[i] = 0.f;
}

__global__ void zero_i_kernel(int* __restrict__ p, int n) {
    int i = blockIdx.x * blockDim.x + threadIdx.x;
    if (i < n) p[i] = 0;
}

// --------------------------------------------------- 1) h = X @ W  (f32 WMMA)
// grid (N/16, F/64), block 128 (4 waves; wave w -> col-tile blockIdx.y*4+w)
__global__ void gemm_h_kernel(const float* __restrict__ X,
                              const float* __restrict__ Wm,
                              float* __restrict__ H) {
    const int lane = threadIdx.x & 31;
    const int wave = threadIdx.x >> 5;
    const int tm   = blockIdx.x;
    const int tn   = blockIdx.y * 4 + wave;
    const int half = lane >> 4;
    const int l16  = lane & 15;
    const int row  = tm * 16 + l16;
    const int col  = tn * 16 + l16;
    v8f c = {};
    for (int k = 0; k < IN_DIM; k += 4) {
        const int kk = k + 2 * half;
        v2f a, b;
        a.x = X[row * IN_DIM + kk];
        a.y = X[row * IN_DIM + kk + 1];
        b.x = Wm[kk * F + col];
        b.y = Wm[(kk + 1) * F + col];
        c = __builtin_amdgcn_wmma_f32_16x16x4_f32(false, a, false, b,
                                                  (short)0, c, false, false);
    }
#pragma unroll
    for (int v = 0; v < 8; ++v)
        H[(size_t)(tm * 16 + v + 8 * half) * F + col] = c[v];
}

// --------------------------------------- 2) per-node attention scores (N*H)
__global__ void attn_score_kernel(const float* __restrict__ H,
                                  const float* __restrict__ al,
                                  const float* __restrict__ ar,
                                  float* __restrict__ esrc,
                                  float* __restrict__ edst) {
    int i = blockIdx.x * blockDim.x + threadIdx.x;
    if (i >= N * HEADS) return;
    const int n = i >> 3, hh = i & 7;
    const float* hp  = H + (size_t)n * F + hh * HID;
    const float* alp = al + hh * HID;
    const float* arp = ar + hh * HID;
    float s0 = 0.f, s1 = 0.f;
    for (int d = 0; d < HID; ++d) { const float hv = hp[d]; s0 += hv * alp[d]; s1 += hv * arp[d]; }
    esrc[i] = s0; edst[i] = s1;
}

// ------------------------------------------------- 3a) CSR: in-degree count
__global__ void deg_count_kernel(const int* __restrict__ dst, int* __restrict__ cnt) {
    int e = blockIdx.x * blockDim.x + threadIdx.x;
    if (e < E) atomicAdd(&cnt[dst[e]], 1);
}

// ------------------------------- 3b) CSR: exclusive scan (single 1024 block)
// thread t owns elements [t*10, t*10+10); LDS Hillis-Steele over 1024 partials
__global__ void scan_kernel(const int* __restrict__ cnt,
                            int* __restrict__ rowstart,
                            int* __restrict__ cursor) {
    __shared__ int sh[1024];
    const int t = threadIdx.x;
    const int base_idx = t * 10;
    int local = 0;
#pragma unroll
    for (int j = 0; j < 10; ++j) {
        const int idx = base_idx + j;
        if (idx < N) local += cnt[idx];
    }
    sh[t] = local;
    __syncthreads();
    for (int off = 1; off < 1024; off <<= 1) {
        const int v = sh[t];
        const int u = (t >= off) ? sh[t - off] : 0;
        __syncthreads();
        sh[t] = v + u;
        __syncthreads();
    }
    int run = (t == 0) ? 0 : sh[t - 1];      // exclusive prefix
#pragma unroll
    for (int j = 0; j < 10; ++j) {
        const int idx = base_idx + j;
        if (idx < N) { rowstart[idx] = run; run += cnt[idx]; cursor[idx] = 0; }
    }
    if (t == 1023) rowstart[N] = sh[1023];
}

// ------------------------------------------------------- 3c) CSR: edge fill
__global__ void fill_kernel(const int* __restrict__ dst,
                            const int* __restrict__ rowstart,
                            int* __restrict__ cursor,
                            int* __restrict__ eidx) {
    int e = blockIdx.x * blockDim.x + threadIdx.x;
    if (e >= E) return;
    const int d = dst[e];
    const int pos = atomicAdd(&cursor[d], 1);
    eidx[rowstart[d] + pos] = e;
}

// ------------- 3d) sort each row ascending (restores determinism; deg ~ 16)
__global__ void sort_rows_kernel(const int* __restrict__ rowstart,
                                 int* __restrict__ eidx) {
    int n = blockIdx.x * blockDim.x + threadIdx.x;
    if (n >= N) return;
    const int s = rowstart[n], e = rowstart[n + 1];
    for (int i = s + 1; i < e; ++i) {        // insertion sort
        const int key = eidx[i];
        int j = i - 1;
        while (j >= s && eidx[j] > key) { eidx[j + 1] = eidx[j]; --j; }
        eidx[j + 1] = key;
    }
}

// --------- 4) per-dst-node GAT aggregation + softmax + ELU (one wave / node)
// lanes: head group hh = lane>>2 (4 lanes per head, HEADS=8);
// feature pass: lane owns features [lane*16, lane*16+16) -> same head mapping.
__global__ void aggregate_kernel(const int* __restrict__ src,
                                 const int* __restrict__ rowstart,
                                 const int* __restrict__ eidx,
                                 const float* __restrict__ esrc,
                                 const float* __restrict__ edst,
                                 const float* __restrict__ H,
                                 float* __restrict__ Zm) {
    const int lane = threadIdx.x & 31;
    const int wave = threadIdx.x >> 5;
    const int n = blockIdx.x * 8 + wave;
    if (n >= N) return;
    const int hh = lane >> 2;
    const int j  = lane & 3;
    const int start = rowstart[n];
    const int deg   = rowstart[n + 1] - start;
    const float edn = edst[n * HEADS + hh];

    // pass 1: per-head max of leaky-relu scores
    float mx = -INFINITY;
    for (int i = j; i < deg; i += 4) {
        const int s = src[eidx[start + i]];
        float v = esrc[s * HEADS + hh] + edn;
        v = v > 0.f ? v : 0.2f * v;
        mx = fmaxf(mx, v);
    }
    mx = fmaxf(mx, __shfl_xor(mx, 1, 32));
    mx = fmaxf(mx, __shfl_xor(mx, 2, 32));

    // pass 2: per-head exp-sum
    float sm = 0.f;
    for (int i = j; i < deg; i += 4) {
        const int s = src[eidx[start + i]];
        float v = esrc[s * HEADS + hh] + edn;
        v = v > 0.f ? v : 0.2f * v;
        sm += __expf(v - mx);
    }
    sm += __shfl_xor(sm, 1, 32);
    sm += __shfl_xor(sm, 2, 32);
    const float inv = 1.f / (sm + 1e-9f);

    // pass 3: accumulate alpha-weighted h[src] rows in registers
    float acc[16];
#pragma unroll
    for (int c = 0; c < 16; ++c) acc[c] = 0.f;
    const int fbase = lane * 16;
    for (int i = 0; i < deg; ++i) {
        const int s = src[eidx[start + i]];
        float v = esrc[s * HEADS + hh] + edn;
        v = v > 0.f ? v : 0.2f * v;
        const float alpha = __expf(v - mx) * inv;
        const float4* hp = (const float4*)(H + (size_t)s * F + fbase);
#pragma unroll
        for (int c4 = 0; c4 < 4; ++c4) {
            const float4 hv = hp[c4];
            acc[c4 * 4 + 0] += hv.x * alpha;
            acc[c4 * 4 + 1] += hv.y * alpha;
            acc[c4 * 4 + 2] += hv.z * alpha;
            acc[c4 * 4 + 3] += hv.w * alpha;
        }
    }
    // fused ELU + writeback
    float* zp = Zm + (size_t)n * F + fbase;
#pragma unroll
    for (int c = 0; c < 16; ++c) {
        const float v = acc[c];
        zp[c] = v > 0.f ? v : (__expf(v) - 1.f);
    }
}

// -------- 5) semantic attention: wsum[m] += sum_n sum_c tanh(z@W1+b1)[n,c]*q[c]
// WMMA GEMM fused with tanh, q-dot and wave reduction. grid (N/16, SEM_HID/16), block 32
__global__ void semantic_kernel(const float* __restrict__ Z,   // [N,F] for metapath m
                                const float* __restrict__ W1,  // [F,SEM_HID]
                                const float* __restrict__ b1,
                                const float* __restrict__ q,
                                float* __restrict__ wsum) {    // scalar accumulator
    const int lane = threadIdx.x & 31;
    const int tm   = blockIdx.x;
    const int tn   = blockIdx.y;
    const int half = lane >> 4;
    const int l16  = lane & 15;
    const int row  = tm * 16 + l16;
    const int col  = tn * 16 + l16;
    v8f c = {};
    for (int k = 0; k < F; k += 4) {
        const int kk = k + 2 * half;
        v2f a, b;
        a.x = Z[(size_t)row * F + kk];
        a.y = Z[(size_t)row * F + kk + 1];
        b.x = W1[kk * SEM_HID + col];
        b.y = W1[(kk + 1) * SEM_HID + col];
        c = __builtin_amdgcn_wmma_f32_16x16x4_f32(false, a, false, b,
                                                  (short)0, c, false, false);
    }
    const float qb = q[col], bb = b1[col];
    float part = 0.f;
#pragma unroll
    for (int v = 0; v < 8; ++v) part += tanhf(c[v] + bb) * qb;
    for (int off = 16; off; off >>= 1) part += __shfl_xor(part, off, 32);
    if (lane == 0) atomicAdd(wsum, part);
}

// ------------------------------------------------- 5b) beta = softmax(w/N)
__global__ void beta_kernel(const float* __restrict__ wsum, float* __restrict__ beta) {
    const float w0 = wsum[0] / N, w1 = wsum[1] / N, w2 = wsum[2] / N;
    const float mx = fmaxf(w0, fmaxf(w1, w2));
    const float e0 = __expf(w0 - mx), e1 = __expf(w1 - mx), e2 = __expf(w2 - mx);
    const float s = e0 + e1 + e2;
    beta[0] = e0 / s; beta[1] = e1 / s; beta[2] = e2 / s;
}

// ------------------- 6) out = (sum_m beta[m] * z[m]) @ fcW + fcb  (f32 WMMA)
// OUT_DIM = 16 -> a single column tile; grid N/16, block 32
__global__ void final_kernel(const float* __restrict__ Z,     // [3][N][F]
                             const float* __restrict__ beta,
                             const float* __restrict__ fcW,   // [F,16]
                             const float* __restrict__ fcb,
                             float* __restrict__ out) {
    const int lane = threadIdx.x & 31;
    const int tm   = blockIdx.x;
    const int half = lane >> 4;
    const int l16  = lane & 15;
    const int row  = tm * 16 + l16;
    const int col  = l16;
    const float b0 = beta[0], b1v = beta[1], b2 = beta[2];
    const float* Z0 = Z;
    const float* Z1 = Z + (size_t)N * F;
    const float* Z2 = Z + 2 * (size_t)N * F;
    v8f c = {};
    for (int k = 0; k < F; k += 4) {
        const int kk = k + 2 * half;
        const size_t r = (size_t)row * F + kk;
        v2f a, b;
        a.x = b0 * Z0[r]     + b1v * Z1[r]     + b2 * Z2[r];
        a.y = b0 * Z0[r + 1] + b1v * Z1[r + 1] + b2 * Z2[r + 1];
        b.x = fcW[kk * OUT_DIM + col];
        b.y = fcW[(kk + 1) * OUT_DIM + col];
        c = __builtin_amdgcn_wmma_f32_16x16x4_f32(false, a, false, b,
                                                  (short)0, c, false, false);
    }
    const float bias = fcb[col];
#pragma unroll
    for (int v = 0; v < 8; ++v)
        out[(size_t)(tm * 16 + v + 8 * half) * OUT_DIM + col] = c[v] + bias;
}

// ---------------------------------------------------------------- launcher

extern "C" void kernel_launch(void* const* d_in, const int* in_sizes, int n_in,
                              void* d_out, int out_size, void* d_ws, size_t ws_size,
                              hipStream_t stream) {
    (void)in_sizes; (void)n_in; (void)out_size; (void)ws_size;
    const float* feat[MP] = {(const float*)d_in[0], (const float*)d_in[3], (const float*)d_in[6]};
    const int*   src[MP]  = {(const int*)d_in[1],   (const int*)d_in[4],   (const int*)d_in[7]};
    const int*   dst[MP]  = {(const int*)d_in[2],   (const int*)d_in[5],   (const int*)d_in[8]};
    const float* W   = (const float*)d_in[9];
    const float* al  = (const float*)d_in[10];
    const float* ar  = (const float*)d_in[11];
    const float* W1  = (const float*)d_in[12];
    const float* b1  = (const float*)d_in[13];
    const float* q   = (const float*)d_in[14];
    const float* fcW = (const float*)d_in[15];
    const float* fcb = (const float*)d_in[16];
    float* out = (float*)d_out;

    // workspace layout: ~84 MB floats + ~0.8 MB ints
    float* ws    = (float*)d_ws;
    float* z     = ws;                              // MP*N*F
    float* h     = z     + (size_t)MP * N * F;      // N*F (reused per metapath)
    float* esrc  = h     + (size_t)N * F;           // N*HEADS
    float* edstb = esrc  + N * HEADS;               // N*HEADS
    float* wsum  = edstb + N * HEADS;               // MP
    float* beta  = wsum  + MP;                      // MP
    int*   cnt      = (int*)(beta + MP);            // N
    int*   rowstart = cnt + N;                      // N+1
    int*   cursor   = rowstart + (N + 1);           // N
    int*   eidx     = cursor + N;                   // E

    zero_f_kernel<<<1, 32, 0, stream>>>(wsum, MP);

    for (int m = 0; m < MP; ++m) {
        // dense feature transform (WMMA f32)
        gemm_h_kernel<<<dim3(N / 16, F / 64), 128, 0, stream>>>(
            feat[m], W + (size_t)m * IN_DIM * F, h);
        attn_score_kernel<<<(N * HEADS + 255) / 256, 256, 0, stream>>>(
            h, al + m * HEADS * HID, ar + m * HEADS * HID, esrc, edstb);
        // CSR by destination
        zero_i_kernel<<<(N + 255) / 256, 256, 0, stream>>>(cnt, N);
        deg_count_kernel<<<(E + 255) / 256, 256, 0, stream>>>(dst[m], cnt);
        scan_kernel<<<1, 1024, 0, stream>>>(cnt, rowstart, cursor);
        fill_kernel<<<(E + 255) / 256, 256, 0, stream>>>(dst[m], rowstart, cursor, eidx);
        sort_rows_kernel<<<(N + 255) / 256, 256, 0, stream>>>(rowstart, eidx);
        // atomic-free softmax + aggregation + fused ELU (one wave per node)
        aggregate_kernel<<<(N + 7) / 8, 256, 0, stream>>>(
            src[m], rowstart, eidx, esrc, edstb, h, z + (size_t)m * N * F);
    }

    for (int m = 0; m < MP; ++m)
        semantic_kernel<<<dim3(N / 16, SEM_HID / 16), 32, 0, stream>>>(
            z + (size_t)m * N * F, W1, b1, q, wsum + m);

    beta_kernel<<<1, 1, 0, stream>>>(wsum, beta);
    final_kernel<<<N / 16, 32, 0, stream>>>(z, beta, fcW, fcb, out);
}